// GCN_82738249990184
// MI455X (gfx1250) — compile-verified
//
#include <hip/hip_runtime.h>
#include <math.h>

// Problem constants
#define BB 4
#define NN 4096
#define FF 64
#define OO 64
#define CHUNKS 64          // partial-sum chunks per batch in k3
#define ROWS_PER_CHUNK (NN / CHUNKS)   // 64

typedef __attribute__((ext_vector_type(2))) float v2f;
typedef __attribute__((ext_vector_type(8))) float v8f;

// ---------------------------------------------------------------------------
// k1: per-row dot products  s_col[r] = x[r,:]·adj_w[:F],  s_row[r] = x[r,:]·adj_w[F:]
// One wave32 per row; lane loads 2 of the 64 floats; shfl-xor tree reduce.
// ---------------------------------------------------------------------------
__global__ void k1_rowdots(const float* __restrict__ x, const float* __restrict__ adj_w,
                           float* __restrict__ s_col, float* __restrict__ s_row) {
    const int lane = threadIdx.x & 31;
    const int wave = threadIdx.x >> 5;
    const int r = blockIdx.x * 8 + wave;            // 0 .. B*N-1
    const float x0 = x[r * FF + lane];
    const float x1 = x[r * FF + lane + 32];
    float sc = x0 * adj_w[lane]      + x1 * adj_w[lane + 32];
    float sr = x0 * adj_w[FF + lane] + x1 * adj_w[FF + lane + 32];
#pragma unroll
    for (int off = 16; off > 0; off >>= 1) {
        sc += __shfl_xor(sc, off, 32);
        sr += __shfl_xor(sr, off, 32);
    }
    if (lane == 0) { s_col[r] = sc; s_row[r] = sr; }
}

// ---------------------------------------------------------------------------
// k2: per-batch deterministic reduction of s_col -> S, then
//     dis[b,i] = (max(N*(s_row+adj_b)+S, 1))^-1/2
// One 1024-thread block per batch.
// ---------------------------------------------------------------------------
__global__ void k2_dis(const float* __restrict__ s_col, const float* __restrict__ s_row,
                       const float* __restrict__ adj_b_p, float* __restrict__ dis) {
    __shared__ float sdata[1024];
    const int b = blockIdx.x;
    const int tid = threadIdx.x;
    const float* sc = s_col + b * NN;
    float local = sc[tid] + sc[tid + 1024] + sc[tid + 2048] + sc[tid + 3072];
    sdata[tid] = local;
    __syncthreads();
    for (int s = 512; s > 0; s >>= 1) {
        if (tid < s) sdata[tid] += sdata[tid + s];
        __syncthreads();
    }
    const float S  = sdata[0];
    const float ab = adj_b_p[0];
#pragma unroll
    for (int k = 0; k < 4; ++k) {
        const int i = tid + k * 1024;
        const float deg = (float)NN * (s_row[b * NN + i] + ab) + S;
        dis[b * NN + i] = 1.0f / sqrtf(fmaxf(deg, 1.0f));
    }
}

// ---------------------------------------------------------------------------
// k3: partial column sums  u[b,f] = sum_n dis*x ,  v[b,f] = sum_n s_col*dis*x
// Grid: B*CHUNKS blocks of 256 threads (4 row-groups x 64 columns).
// ---------------------------------------------------------------------------
__global__ void k3_partials(const float* __restrict__ x, const float* __restrict__ s_col,
                            const float* __restrict__ dis,
                            float* __restrict__ u_part, float* __restrict__ v_part) {
    __shared__ float smu[256], smv[256];
    const int b = blockIdx.x >> 6;          // / CHUNKS
    const int chunk = blockIdx.x & (CHUNKS - 1);
    const int f  = threadIdx.x & 63;
    const int rg = threadIdx.x >> 6;        // 0..3
    const int base = b * NN + chunk * ROWS_PER_CHUNK;
    float au = 0.f, av = 0.f;
    for (int nn = rg; nn < ROWS_PER_CHUNK; nn += 4) {
        const int row = base + nn;
        const float d   = dis[row];
        const float scd = s_col[row] * d;
        const float xv  = x[row * FF + f];
        au += d * xv;
        av += scd * xv;
    }
    smu[threadIdx.x] = au;
    smv[threadIdx.x] = av;
    __syncthreads();
    if (rg == 0) {
        const float tu = smu[f] + smu[f + 64] + smu[f + 128] + smu[f + 192];
        const float tv = smv[f] + smv[f + 64] + smv[f + 128] + smv[f + 192];
        u_part[(b * CHUNKS + chunk) * FF + f] = tu;
        v_part[(b * CHUNKS + chunk) * FF + f] = tv;
    }
}

// ---------------------------------------------------------------------------
// k4: combine partials into zero-padded stacked A matrix uv (16 x 64):
//     rows 0-3 = u[b], rows 4-7 = v[b], rows 8-15 = 0. Fixed-order sums.
// ---------------------------------------------------------------------------
__global__ void k4_combine(const float* __restrict__ u_part, const float* __restrict__ v_part,
                           float* __restrict__ uv) {
    const int tid = threadIdx.x; // 256 threads
#pragma unroll
    for (int k = 0; k < 4; ++k) {
        const int e = tid + k * 256;          // 0..1023
        const int row = e >> 6;
        const int f = e & 63;
        float val = 0.f;
        if (row < 4) {
            const int b = row;
            for (int c = 0; c < CHUNKS; ++c) val += u_part[(b * CHUNKS + c) * FF + f];
        } else if (row < 8) {
            const int b = row - 4;
            for (int c = 0; c < CHUNKS; ++c) val += v_part[(b * CHUNKS + c) * FF + f];
        }
        uv[e] = val;
    }
}

// ---------------------------------------------------------------------------
// k5: WMMA GEMM  PQ(16x64) = uv(16x64) @ W(64x64), fp32 via V_WMMA_F32_16X16X4_F32.
// One wave. A layout: vgpr0/1 hold K={0,1} for lanes 0-15, K={2,3} for lanes 16-31.
// B layout mirrors it (rows striped across lanes). C rows 0-7 live in lanes 0-15.
// ---------------------------------------------------------------------------
__global__ void k5_wmma(const float* __restrict__ uv, const float* __restrict__ W,
                        float* __restrict__ PQ) {
    const int lane = threadIdx.x;       // 0..31
    const int m  = lane & 15;
    const int hi = (lane >> 4) << 1;    // 0 or 2
#pragma unroll
    for (int t = 0; t < 4; ++t) {       // O tiles of 16
        v8f acc = {};
#pragma unroll
        for (int k0 = 0; k0 < 16; ++k0) {   // K = 64 in steps of 4
            const int kb = 4 * k0 + hi;
            v2f a, bm;
            a.x  = uv[m * FF + kb];
            a.y  = uv[m * FF + kb + 1];
            const int n = t * 16 + m;
            bm.x = W[kb * OO + n];
            bm.y = W[(kb + 1) * OO + n];
            acc = __builtin_amdgcn_wmma_f32_16x16x4_f32(
                /*neg_a=*/false, a, /*neg_b=*/false, bm,
                /*c_mod=*/(short)0, acc, /*reuse_a=*/false, /*reuse_b=*/false);
        }
        if (lane < 16) {
#pragma unroll
            for (int r = 0; r < 8; ++r)       // rows 0-7 = P (0-3) and Q (4-7)
                PQ[r * OO + t * 16 + lane] = acc[r];
        }
    }
}

// ---------------------------------------------------------------------------
// k6: epilogue  out[b,i,o] = relu( dis*( (s_row+adj_b)*P[b,o] + Q[b,o] ) + bias[o] )
// ---------------------------------------------------------------------------
__global__ void k6_out(const float* __restrict__ dis, const float* __restrict__ s_row,
                       const float* __restrict__ PQ, const float* __restrict__ bias,
                       const float* __restrict__ adj_b_p, float* __restrict__ out) {
    const int r = blockIdx.x * 4 + (threadIdx.x >> 6);   // row 0..B*N-1
    const int o = threadIdx.x & 63;
    const int b = r >> 12;                                // / NN
    const float d   = dis[r];
    const float srb = s_row[r] + adj_b_p[0];
    const float P = PQ[b * OO + o];
    const float Q = PQ[(4 + b) * OO + o];
    const float val = d * (srb * P + Q) + bias[o];
    out[r * OO + o] = fmaxf(val, 0.0f);
}

extern "C" void kernel_launch(void* const* d_in, const int* in_sizes, int n_in,
                              void* d_out, int out_size, void* d_ws, size_t ws_size,
                              hipStream_t stream) {
    const float* x      = (const float*)d_in[0];  // (B,N,F)
    const float* adj_w  = (const float*)d_in[1];  // (2F,)
    const float* adj_b  = (const float*)d_in[2];  // scalar
    const float* weight = (const float*)d_in[3];  // (F,O)
    const float* bias   = (const float*)d_in[4];  // (O,)
    float* out = (float*)d_out;
    float* ws  = (float*)d_ws;

    // workspace layout (floats)
    float* s_col  = ws;                         // 16384
    float* s_row  = ws + 16384;                 // 16384
    float* dis    = ws + 32768;                 // 16384
    float* u_part = ws + 49152;                 // B*CHUNKS*F = 16384
    float* v_part = ws + 65536;                 // 16384
    float* uv     = ws + 81920;                 // 16*64 = 1024
    float* PQ     = ws + 82944;                 // 8*64  = 512
    // total: 83456 floats (~334 KB)

    k1_rowdots<<<(BB * NN) / 8, 256, 0, stream>>>(x, adj_w, s_col, s_row);
    k2_dis    <<<BB, 1024, 0, stream>>>(s_col, s_row, adj_b, dis);
    k3_partials<<<BB * CHUNKS, 256, 0, stream>>>(x, s_col, dis, u_part, v_part);
    k4_combine<<<1, 256, 0, stream>>>(u_part, v_part, uv);
    k5_wmma   <<<1, 32, 0, stream>>>(uv, weight, PQ);
    k6_out    <<<(BB * NN) / 4, 256, 0, stream>>>(dis, s_row, PQ, bias, adj_b, out);
}